// HyperbolicGraphAttentionLayer_21938692948001
// MI455X (gfx1250) — compile-verified
//
#include <hip/hip_runtime.h>
#include <hip/hip_bf16.h>
#include <math.h>

// Hyperbolic graph attention layer for MI455X (gfx1250, wave32).
// D_IN = D_OUT = 64 hardcoded (per reference). All math fp32.
//
// Workspace layout (floats), requires ws_size >= 4*N*64*4 bytes (~102 MB):
//   [0,        N*64)  : p      -> reused as att_log_h accumulator
//   [N*64,   2*N*64)  : Mp     -> reused as e[E] | e_max_bits[N] | denom[N]
//   [2*N*64, 3*N*64)  : hh
//   [3*N*64, 4*N*64)  : log_h

#define EPSV  1e-15f
#define CLIPV 0.98f

typedef __attribute__((ext_vector_type(2))) float v2f;
typedef __attribute__((ext_vector_type(8))) float v8f;

__device__ __forceinline__ float wsum(float v) {
#pragma unroll
  for (int o = 16; o > 0; o >>= 1) v += __shfl_xor(v, o, 32);
  return v;
}

// ---- Stage 1+2a: f2p projection, then +EPS and clip (pre-GEMM p) ----
// One wave32 per node row; lane owns features {lane, lane+32}.
__global__ void k_project_p(const float* __restrict__ h, float* __restrict__ p, int N) {
  const int wave = threadIdx.x >> 5, lane = threadIdx.x & 31;
  const int row = blockIdx.x * 8 + wave;
  if (row >= N) return;
  const float* hr = h + (size_t)row * 64;
  float x0 = hr[lane] + EPSV;
  float x1 = hr[lane + 32] + EPSV;
  float fn0 = sqrtf(wsum(x0 * x0 + x1 * x1));
  float s = fn0 > CLIPV ? CLIPV / fn0 : 1.0f;
  x0 *= s; x1 *= s;
  float fn = fn0 * s;
  float fac = tanhf(fn) / fn;           // p = tanh(|f|)/|f| * f
  float q0 = fac * x0 + EPSV;           // p2h pre-step: p += EPS
  float q1 = fac * x1 + EPSV;
  float qn = sqrtf(wsum(q0 * q0 + q1 * q1));
  float s2 = qn > CLIPV ? CLIPV / qn : 1.0f;   // clip
  float* pr = p + (size_t)row * 64;
  pr[lane]      = q0 * s2;
  pr[lane + 32] = q1 * s2;
}

// ---- Stage 2b: Mp = p @ W via V_WMMA_F32_16X16X4_F32 (exact fp32) ----
// Block = 4 waves; wave w computes the 16x16 tile at (blockIdx.x*16, w*16).
// K = 64 -> 16 chained WMMAs. EXEC must stay all-ones: loads are clamped,
// only the stores are predicated.
__global__ __launch_bounds__(128) void k_gemm_wmma(const float* __restrict__ P,
                                                   const float* __restrict__ W,
                                                   float* __restrict__ Mp, int N) {
  const int lane = threadIdx.x & 31;
  const int wave = threadIdx.x >> 5;   // 0..3 -> column tile
  const int half = lane >> 4;          // 0/1
  const int l16  = lane & 15;
  const int row0 = blockIdx.x * 16;
  const int col0 = wave * 16;

  int mrow = row0 + l16;
  if (mrow >= N) mrow = N - 1;         // clamp, keep EXEC full

  v8f acc = {};
#pragma unroll
  for (int kb = 0; kb < 64; kb += 4) {
    const int k0 = kb + 2 * half;
    v2f a, b;
    // A 16x4 fp32 fragment: lanes 0-15 hold K={kb,kb+1}, lanes 16-31 K={kb+2,kb+3}
    a.x = P[(size_t)mrow * 64 + k0 + 0];
    a.y = P[(size_t)mrow * 64 + k0 + 1];
    // B 4x16 fp32 fragment (mirrored): vgpr v holds K = kb + v + 2*half, N = l16
    b.x = W[(size_t)(k0 + 0) * 64 + col0 + l16];
    b.y = W[(size_t)(k0 + 1) * 64 + col0 + l16];
    acc = __builtin_amdgcn_wmma_f32_16x16x4_f32(
        /*neg_a=*/false, a, /*neg_b=*/false, b,
        /*c_mod=*/(short)0, acc, /*reuse_a=*/false, /*reuse_b=*/false);
  }
#pragma unroll
  for (int v = 0; v < 8; ++v) {
    const int m = row0 + v + 8 * half; // C/D layout: vgpr v, half selects M or M+8
    if (m < N) Mp[(size_t)m * 64 + col0 + l16] = acc[v];
  }
}

// ---- Stage 2c: p2h epilogue -> hh ; fused log-projection -> log_h ----
__global__ void k_p2h(const float* __restrict__ p, const float* __restrict__ Mp,
                      const float* __restrict__ cptr,
                      float* __restrict__ hh, float* __restrict__ logh, int N) {
  const int wave = threadIdx.x >> 5, lane = threadIdx.x & 31;
  const int row = blockIdx.x * 8 + wave;
  if (row >= N) return;
  const float sc = sqrtf(cptr[0]);
  const float* pr = p + (size_t)row * 64;
  float p0 = pr[lane], p1 = pr[lane + 32];
  float pn = sqrtf(wsum(p0 * p0 + p1 * p1));
  float an = atanhf(fminf(sc * pn, 0.9f));
  const float* mr = Mp + (size_t)row * 64;
  float t0 = mr[lane]      * an / pn / sc;
  float t1 = mr[lane + 32] * an / pn / sc;
  const float clip2 = CLIPV / sc;
  float tn0 = sqrtf(wsum(t0 * t0 + t1 * t1));
  float s = tn0 > clip2 ? clip2 / tn0 : 1.0f;
  t0 *= s; t1 *= s;
  float tn = tn0 * s;
  float fac = tanhf(sc * tn) / (sc * tn);
  float h0 = fac * t0, h1 = fac * t1;
  float* hr = hh + (size_t)row * 64;
  hr[lane] = h0; hr[lane + 32] = h1;
  // log projection of hh
  float g0 = h0 + EPSV, g1 = h1 + EPSV;
  float gn0 = sqrtf(wsum(g0 * g0 + g1 * g1));
  float s3 = gn0 > CLIPV ? CLIPV / gn0 : 1.0f;
  g0 *= s3; g1 *= s3;
  float gn = gn0 * s3;
  float lf = atanhf(fminf(sc * gn, 0.9f)) / gn;
  float* lr = logh + (size_t)row * 64;
  lr[lane] = lf * g0; lr[lane + 32] = lf * g1;
}

// ---- init accumulators ----
__global__ void k_init(float* __restrict__ att, unsigned* __restrict__ emax,
                       float* __restrict__ denom, int N) {
  const int i = blockIdx.x * blockDim.x + threadIdx.x;
  if (i < N * 64) att[i] = 0.0f;
  if (i < N) { emax[i] = 0xFF800000u; denom[i] = 0.0f; }  // -inf bits
}

// ---- Stage 3: per-edge Mobius distance + segment max (wave32 per edge) ----
// All e < 0 (e = -2*atanh(clip(.,1e-8,.98))), so float-max == u32-min on bits.
__global__ void k_edge_dist(const float* __restrict__ hh, const int* __restrict__ ei,
                            const float* __restrict__ cptr,
                            float* __restrict__ earr, unsigned* __restrict__ emax, int E) {
  const int wave = threadIdx.x >> 5, lane = threadIdx.x & 31;
  const long long e = (long long)blockIdx.x * 8 + wave;
  if (e >= E) return;
  const int r = ei[e], cidx = ei[(size_t)E + e];
  const float* ur = hh + (size_t)r * 64;
  const float* vr = hh + (size_t)cidx * 64;
  float u0 = -ur[lane], u1 = -ur[lane + 32];
  float v0 =  vr[lane], v1 =  vr[lane + 32];
  float nu  = wsum(u0 * u0 + u1 * u1);
  float nv  = wsum(v0 * v0 + v1 * v1);
  float duv = wsum(u0 * v0 + u1 * v1);
  const float c0 = cptr[0], sc = sqrtf(c0);
  float coef1 = 1.0f + 2.0f * c0 * duv + c0 * nv;
  float coef2 = 1.0f - c0 * nu;
  float den   = 1.0f + 2.0f * c0 * duv + c0 * c0 * nu * nv;
  float m0 = (coef1 * u0 + coef2 * v0) / den;
  float m1 = (coef1 * u1 + coef2 * v1) / den;
  float rn = sqrtf(wsum(m0 * m0 + m1 * m1));
  float arg = fminf(fmaxf(sc * rn, 1e-8f), CLIPV);
  float ev = (-2.0f / sc) * atanhf(arg);
  if (lane == 0) {
    earr[e] = ev;
    atomicMin(&emax[r], __float_as_uint(ev));
  }
}

// ---- Stage 4a: exp(e - max) and segment sum ----
__global__ void k_softmax_num(const int* __restrict__ ei, float* __restrict__ earr,
                              const unsigned* __restrict__ emax,
                              float* __restrict__ denom, int E) {
  const long long i = (long long)blockIdx.x * blockDim.x + threadIdx.x;
  if (i >= E) return;
  const int r = ei[i];
  float ex = expf(earr[i] - __uint_as_float(emax[r]));
  earr[i] = ex;
  atomicAdd(&denom[r], ex);
}

// ---- Stage 4b+6: attention * log_h[col] scatter-add (wave32 per edge) ----
__global__ void k_aggregate(const int* __restrict__ ei, const float* __restrict__ earr,
                            const float* __restrict__ denom, const float* __restrict__ logh,
                            const float* __restrict__ cptr, float* __restrict__ att, int E) {
  const int wave = threadIdx.x >> 5, lane = threadIdx.x & 31;
  const long long e = (long long)blockIdx.x * 8 + wave;
  if (e >= E) return;
  const int r = ei[e], cidx = ei[(size_t)E + e];
  const float sc = sqrtf(cptr[0]);
  const float w = sc * (earr[e] / (denom[r] + 1e-16f));
  const float* lr = logh + (size_t)cidx * 64;
  atomicAdd(&att[(size_t)r * 64 + lane],      w * lr[lane]);
  atomicAdd(&att[(size_t)r * 64 + lane + 32], w * lr[lane + 32]);
}

// ---- Stage 7: elu + exp projection -> h_prime ----
__global__ void k_final(const float* __restrict__ att, const float* __restrict__ cptr,
                        float* __restrict__ out, int N) {
  const int wave = threadIdx.x >> 5, lane = threadIdx.x & 31;
  const int row = blockIdx.x * 8 + wave;
  if (row >= N) return;
  const float sc = sqrtf(cptr[0]);
  const float* ar = att + (size_t)row * 64;
  float z0 = ar[lane], z1 = ar[lane + 32];
  z0 = (z0 > 0.0f ? z0 : expm1f(z0)) + EPSV;   // elu + EPS
  z1 = (z1 > 0.0f ? z1 : expm1f(z1)) + EPSV;
  const float clip3 = CLIPV / sc;
  float zn0 = sqrtf(wsum(z0 * z0 + z1 * z1));
  float s = zn0 > clip3 ? clip3 / zn0 : 1.0f;
  z0 *= s; z1 *= s;
  float zn = zn0 * s;
  float fac = tanhf(sc * zn) / (sc * zn);
  float* o = out + (size_t)row * 64;
  o[lane] = fac * z0; o[lane + 32] = fac * z1;
}

extern "C" void kernel_launch(void* const* d_in, const int* in_sizes, int n_in,
                              void* d_out, int out_size, void* d_ws, size_t ws_size,
                              hipStream_t stream) {
  const float* h  = (const float*)d_in[0];
  const float* W  = (const float*)d_in[1];
  const float* c  = (const float*)d_in[2];
  const int*   ei = (const int*)d_in[3];
  const int N = in_sizes[0] / 64;
  const int E = in_sizes[3] / 2;

  float* ws = (float*)d_ws;
  const size_t nodeF = (size_t)N * 64;
  float* p    = ws;                 // later reused as att_log_h
  float* mp   = ws + nodeF;         // later reused: e[E] | emax[N] | denom[N]
  float* hh   = ws + 2 * nodeF;
  float* logh = ws + 3 * nodeF;
  float*    earr  = mp;
  unsigned* emax  = (unsigned*)(mp + E);
  float*    denom = mp + E + N;
  float*    att   = p;
  float*    out   = (float*)d_out;

  const int rowBlocks  = (N + 7) / 8;
  const int tileBlocks = (N + 15) / 16;
  const int edgeWB     = (E + 7) / 8;

  k_project_p <<<rowBlocks, 256, 0, stream>>>(h, p, N);
  k_gemm_wmma <<<tileBlocks, 128, 0, stream>>>(p, W, mp, N);
  k_p2h       <<<rowBlocks, 256, 0, stream>>>(p, mp, c, hh, logh, N);
  k_init      <<<(int)((nodeF + 255) / 256), 256, 0, stream>>>(att, emax, denom, N);
  k_edge_dist <<<edgeWB, 256, 0, stream>>>(hh, ei, c, earr, emax, E);
  k_softmax_num<<<(E + 255) / 256, 256, 0, stream>>>(ei, earr, emax, denom, E);
  k_aggregate <<<edgeWB, 256, 0, stream>>>(ei, earr, denom, logh, c, att, E);
  k_final     <<<rowBlocks, 256, 0, stream>>>(att, c, out, N);
}